// KroneckerLoss_2302102471322
// MI455X (gfx1250) — compile-verified
//
#include <hip/hip_runtime.h>

// KroneckerLoss on MI455X (gfx1250, wave32).
// loss = sum_b 0.5*((a*h1*c - t1)^2 + (a*h2*c + t1)^2)
//   a  = sum_k obs1[b,k],  h1/h2 = sum_k hyp1/hyp2[b,k]
//   c  = sum_k obs2[b,k] * (-1)^k
// Memory-bound: 128 MiB @ 23.3 TB/s ~ 5.8us. Row reductions are done with
// V_WMMA_F32_16X16X4_F32 (C += A(16x4) x ones(4x16)); the alternating sign is
// folded into A's odd-K element so the B operand layout is irrelevant.

typedef __attribute__((ext_vector_type(2))) float v2f;
typedef __attribute__((ext_vector_type(8))) float v8f;

#define NB 8192
#define DIM 1024
#define ROWS_PER_BLOCK 16
#define NBLOCKS (NB / ROWS_PER_BLOCK)   // 512

__global__ __launch_bounds__(256)
void kron_loss_stage1(const float* __restrict__ obs1,
                      const float* __restrict__ obs2,
                      const float* __restrict__ hyp1,
                      const float* __restrict__ hyp2,
                      const int*   __restrict__ labels,
                      float*       __restrict__ partials)
{
    __shared__ float part[4][2][16];   // [matrix][k-half][row]
    __shared__ float ls[16];           // per-sample losses

    const int lane = threadIdx.x & 31;
    const int wave = threadIdx.x >> 5;   // 0..7
    const int mat  = wave >> 1;          // 0..3: obs1, obs2, hyp1, hyp2
    const int half = wave & 1;           // k in [half*512, half*512+512)
    const int r0   = blockIdx.x * ROWS_PER_BLOCK;

    const float* X = (mat == 0) ? obs1 : (mat == 1) ? obs2
                   : (mat == 2) ? hyp1 : hyp2;

    // WMMA 16x4 f32 A-layout: lanes 0-15 -> (row=lane, K=0/1 in vgpr0/1),
    // lanes 16-31 -> (row=lane-16, K=2/3). One global_load_b64 per lane per step.
    const int row  = lane & 15;
    const int koff = (lane >> 4) * 2;
    const float* rp = X + (size_t)(r0 + row) * DIM + half * (DIM / 2) + koff;

    // alternating sign: A element 0 is always even-K, element 1 always odd-K
    const float sgn = (mat == 1) ? -1.0f : 1.0f;
    const v2f bones = {1.0f, 1.0f};      // B = ones -> every C column = row sum

    v8f acc = {};
    #pragma unroll 8
    for (int k = 0; k < DIM / 2; k += 4) {
        v2f a = __builtin_nontemporal_load((const v2f*)(rp + k));
        a.y *= sgn;
        acc = __builtin_amdgcn_wmma_f32_16x16x4_f32(
                  /*neg_a=*/false, a, /*neg_b=*/false, bones,
                  /*c_mod=*/(short)0, acc, /*reuse_a=*/false, /*reuse_b=*/false);
    }

    // C[m,n] identical over n. Lane with (lane&15)==0 holds rows
    // j + 8*(lane>=16) in acc[j]  (C/D layout: vgpr j -> M=j / M=j+8).
    if ((lane & 15) == 0) {
        const int rbase = (lane >> 4) * 8;
        #pragma unroll
        for (int j = 0; j < 8; ++j)
            part[mat][half][rbase + j] = acc[j];
    }
    __syncthreads();

    if (threadIdx.x < 16) {
        const int m = threadIdx.x;
        const float a  = part[0][0][m] + part[0][1][m];
        const float c  = part[1][0][m] + part[1][1][m];
        const float h1 = part[2][0][m] + part[2][1][m];
        const float h2 = part[3][0][m] + part[3][1][m];
        const float t1 = (labels[r0 + m] == 1) ? 1.0f : -1.0f;  // t2 = -t1
        const float d1 = a * h1 * c - t1;
        const float d2 = a * h2 * c + t1;
        ls[m] = 0.5f * (d1 * d1 + d2 * d2);
    }
    __syncthreads();

    if (threadIdx.x == 0) {
        float s = 0.0f;
        #pragma unroll
        for (int j = 0; j < 16; ++j) s += ls[j];
        partials[blockIdx.x] = s;   // deterministic: no global atomics
    }
}

__global__ __launch_bounds__(256)
void kron_loss_stage2(const float* __restrict__ partials,
                      float*       __restrict__ out, int n)
{
    __shared__ float sm[256];
    float s = 0.0f;
    for (int i = threadIdx.x; i < n; i += 256)
        s += partials[i];
    sm[threadIdx.x] = s;
    __syncthreads();
    #pragma unroll
    for (int off = 128; off > 0; off >>= 1) {
        if ((int)threadIdx.x < off) sm[threadIdx.x] += sm[threadIdx.x + off];
        __syncthreads();
    }
    if (threadIdx.x == 0) out[0] = sm[0];
}

extern "C" void kernel_launch(void* const* d_in, const int* in_sizes, int n_in,
                              void* d_out, int out_size, void* d_ws, size_t ws_size,
                              hipStream_t stream)
{
    const float* obs1   = (const float*)d_in[0];
    const float* obs2   = (const float*)d_in[1];
    const float* hyp1   = (const float*)d_in[2];
    const float* hyp2   = (const float*)d_in[3];
    const int*   labels = (const int*)d_in[4];
    float* partials = (float*)d_ws;     // NBLOCKS floats of scratch

    kron_loss_stage1<<<NBLOCKS, 256, 0, stream>>>(obs1, obs2, hyp1, hyp2,
                                                  labels, partials);
    kron_loss_stage2<<<1, 256, 0, stream>>>(partials, (float*)d_out, NBLOCKS);
}